// MeshUnpool_31336081392112
// MI455X (gfx1250) — compile-verified
//
#include <hip/hip_runtime.h>
#include <stdint.h>
#include <stddef.h>

// Problem constants (from the reference): B=16, NF=256, E=12000, U=24000, NNZ=48000
static constexpr int kB    = 16;
static constexpr int kNF   = 256;
static constexpr int kE    = 12000;
static constexpr int kU    = 24000;
static constexpr int kNNZ  = 48000;
static constexpr int kThreads = 256;
static constexpr int kNNZ4 = kNNZ / 4;   // 12000 vec4 groups, exact
static constexpr int kU4   = kU / 4;     // 6000 vec4 groups, exact
// LDS: staged feature row (E floats) + accumulated output row (U floats) = 144000 B
static constexpr unsigned kSmemBytes = (unsigned)((kE + kU) * sizeof(float));

typedef unsigned int v4u __attribute__((ext_vector_type(4)));
typedef int          v8i __attribute__((ext_vector_type(8)));
typedef int          v4i __attribute__((ext_vector_type(4)));

// ---- Tensor Data Mover descriptor helpers (CDNA5 ISA) ----
// D# group 0: [1:0]=count(1=valid) | [63:32]=lds_addr | [120:64]=global_addr | [127:126]=type(2)
__device__ __forceinline__ v4u tdm_group0(uint32_t lds_off, const void* gptr) {
  uint64_t ga = (uint64_t)(uintptr_t)gptr;
  v4u g0;
  g0.x = 1u;                                                // count=1, user mode, no gather
  g0.y = lds_off;                                           // LDS byte address
  g0.z = (uint32_t)ga;                                      // global_addr[31:0]
  g0.w = ((uint32_t)(ga >> 32) & 0x01FFFFFFu) | (2u << 30); // global_addr[56:32] | type=2
  return g0;
}

// D# group 1 for a simple 1-D contiguous tile of `ndw` dwords (data_size=4B):
//   [17:16]=data_size(2) ; tensor_dim0=ndw ; tensor_dim1=1 ; tile_dim0=ndw ;
//   tile_dim1=1 ; tile_dim2=0 ; tensor_dim0_stride=ndw ; no pad, no multicast.
__device__ __forceinline__ v8i tdm_group1_1d(uint32_t ndw) {
  v8i g1;
  g1[0] = (int)(2u << 16);                                  // data_size = 4 bytes
  g1[1] = (int)((ndw & 0xFFFFu) << 16);                     // tensor_dim0[15:0] @ bits 63:48
  g1[2] = (int)(((ndw >> 16) & 0xFFFFu) | (1u << 16));      // tensor_dim0[31:16] | tensor_dim1=1
  g1[3] = (int)((ndw & 0xFFFFu) << 16);                     // tensor_dim1 hi=0 | tile_dim0
  g1[4] = 1;                                                // tile_dim1=1, tile_dim2=0
  g1[5] = (int)ndw;                                         // tensor_dim0_stride[31:0]
  g1[6] = 0;                                                // stride0 hi | stride1 lo
  g1[7] = 0;                                                // stride1 hi
  return g1;
}

__device__ __forceinline__ void tdm_load_1d(uint32_t lds_off, const void* g, uint32_t ndw) {
  v4i z4 = {0, 0, 0, 0};
  v8i z8 = {0, 0, 0, 0, 0, 0, 0, 0};
  __builtin_amdgcn_tensor_load_to_lds(tdm_group0(lds_off, g), tdm_group1_1d(ndw),
                                      z4, z4, z8, 0);
}

__device__ __forceinline__ void tdm_store_1d(uint32_t lds_off, void* g, uint32_t ndw) {
  v4i z4 = {0, 0, 0, 0};
  v8i z8 = {0, 0, 0, 0, 0, 0, 0, 0};
  __builtin_amdgcn_tensor_store_from_lds(tdm_group0(lds_off, g), tdm_group1_1d(ndw),
                                         z4, z4, z8, 0);
}

// ---- Kernel 0: w[b,n] = values[b,n] / occurrences[b, cols[b,n]] ----
__global__ __launch_bounds__(kThreads)
void precompute_w_kernel(const float* __restrict__ values,
                         const float* __restrict__ occ,
                         const int*   __restrict__ cols,
                         float*       __restrict__ w) {
  int i = blockIdx.x * kThreads + threadIdx.x;
  if (i < kB * kNNZ) {
    int b = i / kNNZ;
    int c = cols[i];
    w[i] = values[i] / occ[(size_t)b * kU + c];
  }
}

// ---- Main kernel: one block per (b, f). Stage feature row via TDM, accumulate
// output row entirely in LDS with ds_add_f32, write back via TDM store. ----
template <bool HAVE_W>
__global__ __launch_bounds__(kThreads)
void mesh_unpool_kernel(const float* __restrict__ features,   // [B, NF, E]
                        const int*   __restrict__ rows,       // [B, NNZ]
                        const int*   __restrict__ cols,       // [B, NNZ]
                        const float* __restrict__ w,          // [B, NNZ] (if HAVE_W)
                        const float* __restrict__ values,     // [B, NNZ] (if !HAVE_W)
                        const float* __restrict__ occ,        // [B, U]   (if !HAVE_W)
                        float*       __restrict__ out) {      // [B, NF, U]
  const int f = blockIdx.x;
  const int b = blockIdx.y;

  extern __shared__ float smem[];
  float* sfeat = smem;        // kE floats: features[b, f, :]
  float* sout  = smem + kE;   // kU floats: out[b, f, :] accumulator

  const uint32_t feat_off = (uint32_t)(uintptr_t)(void*)sfeat;
  const uint32_t out_off  = (uint32_t)(uintptr_t)(void*)sout;

  const float* gfeat = features + ((size_t)b * kNF + f) * kE;
  float*       gout  = out      + ((size_t)b * kNF + f) * kU;

  // Wave 0 kicks off the TDM load of the 48 KB feature row (TENSORcnt-tracked,
  // EXEC-independent) while every thread zeroes the LDS output row (b128 stores).
  if (threadIdx.x < 32) {
    tdm_load_1d(feat_off, gfeat, (uint32_t)kE);
  }
  {
    float4* sout4 = reinterpret_cast<float4*>(sout);
    const float4 z4 = make_float4(0.f, 0.f, 0.f, 0.f);
    for (int i = threadIdx.x; i < kU4; i += kThreads) sout4[i] = z4;
  }
  if (threadIdx.x < 32) {
    __builtin_amdgcn_s_wait_tensorcnt(0);
  }
  __syncthreads();

  // Gather / scale / LDS scatter-add over all nonzeros of this mesh.
  // Metadata is consumed 4 edges at a time via b128 loads (48000 % 4 == 0).
  const size_t base = (size_t)b * kNNZ;
  const int4*   rows4 = reinterpret_cast<const int4*>(rows + base);
  const int4*   cols4 = reinterpret_cast<const int4*>(cols + base);
  const float4* w4    = HAVE_W ? reinterpret_cast<const float4*>(w + base) : nullptr;
  const float4* val4  = HAVE_W ? nullptr : reinterpret_cast<const float4*>(values + base);

  for (int i = threadIdx.x; i < kNNZ4; i += kThreads) {
    const int4 r = rows4[i];
    const int4 c = cols4[i];
    float4 v;
    if constexpr (HAVE_W) {
      v = w4[i];
    } else {
      const float4 vraw = val4[i];
      const float* ob = occ + (size_t)b * kU;
      v.x = vraw.x / ob[c.x];
      v.y = vraw.y / ob[c.y];
      v.z = vraw.z / ob[c.z];
      v.w = vraw.w / ob[c.w];
    }
    atomicAdd(&sout[c.x], sfeat[r.x] * v.x);   // ds_add_f32, no global atomics
    atomicAdd(&sout[c.y], sfeat[r.y] * v.y);
    atomicAdd(&sout[c.z], sfeat[r.z] * v.z);
    atomicAdd(&sout[c.w], sfeat[r.w] * v.w);
  }
  __syncthreads();

  // Wave 0 DMAs the completed 96 KB output row back to HBM.
  if (threadIdx.x < 32) {
    tdm_store_1d(out_off, gout, (uint32_t)kU);
    __builtin_amdgcn_s_wait_tensorcnt(0);
  }
}

extern "C" void kernel_launch(void* const* d_in, const int* in_sizes, int n_in,
                              void* d_out, int out_size, void* d_ws, size_t ws_size,
                              hipStream_t stream) {
  const float* features = (const float*)d_in[0];   // [B, NF, E]
  const float* values   = (const float*)d_in[1];   // [B, NNZ]
  const float* occ      = (const float*)d_in[2];   // [B, U]
  const int*   rows     = (const int*)  d_in[3];   // [B, NNZ]
  const int*   cols     = (const int*)  d_in[4];   // [B, NNZ]
  float*       out      = (float*)d_out;           // [B, NF, U]
  (void)in_sizes; (void)n_in; (void)out_size;

  const size_t w_bytes = (size_t)kB * kNNZ * sizeof(float);
  const bool use_ws = (d_ws != nullptr) && (ws_size >= w_bytes);

  // 144 KB dynamic LDS per block (> default 64 KB cap) — raise the limit.
  (void)hipFuncSetAttribute((const void*)mesh_unpool_kernel<true>,
                            hipFuncAttributeMaxDynamicSharedMemorySize, (int)kSmemBytes);
  (void)hipFuncSetAttribute((const void*)mesh_unpool_kernel<false>,
                            hipFuncAttributeMaxDynamicSharedMemorySize, (int)kSmemBytes);

  dim3 grid(kNF, kB);
  dim3 block(kThreads);

  if (use_ws) {
    float* w = (float*)d_ws;
    int total = kB * kNNZ;
    precompute_w_kernel<<<(total + kThreads - 1) / kThreads, kThreads, 0, stream>>>(
        values, occ, cols, w);
    mesh_unpool_kernel<true><<<grid, block, kSmemBytes, stream>>>(
        features, rows, cols, w, nullptr, nullptr, out);
  } else {
    mesh_unpool_kernel<false><<<grid, block, kSmemBytes, stream>>>(
        features, rows, cols, nullptr, values, occ, out);
  }
}